// KPConvLayer_90056874262833
// MI455X (gfx1250) — compile-verified
//
#include <hip/hip_runtime.h>

typedef float v2f __attribute__((ext_vector_type(2)));
typedef float v8f __attribute__((ext_vector_type(8)));

#define KP     15
#define NNB    32
#define FDIM   64
#define CDIM   64
#define TILE_M 16
#define WAVES  4
#define WSTR   (NNB * 16)        // per-wave W tile: [n=32][k=16]
#define WF_STRIDE 964            // 15*64=960 + 4 pad -> lane-strided reads hit distinct banks

__global__ __launch_bounds__(128) void kpconv_wmma_kernel(
    const float* __restrict__ points,       // [N,3]
    const float* __restrict__ features,     // [N,64]
    const float* __restrict__ out_points,   // [M,3]
    const int*   __restrict__ nbr_idx,      // [M,32]
    const float* __restrict__ k_points,     // [15,3]
    const float* __restrict__ k_values,     // [15,64,64] == flat [960,64]
    float* __restrict__ out,                // [M,64]
    int M)
{
    __shared__ float W_lds[WAVES * WSTR];          // 8 KB   per-wave weight tile
    __shared__ float WF_lds[TILE_M * WF_STRIDE];   // ~60 KB intermediate WF, fp32
    __shared__ int   idx_lds[WAVES * NNB];         // per-wave neighbor indices
    __shared__ float q_lds[TILE_M * 3];
    __shared__ float kp_lds[KP * 3];

    const int tid  = threadIdx.x;
    const int wave = tid >> 5;
    const int lane = tid & 31;
    const int half = lane >> 4;     // 0: lanes 0-15, 1: lanes 16-31
    const int l    = lane & 15;

    const int m0 = blockIdx.x * TILE_M;

    // stage output-point coords + kernel-point coords
    if (tid < TILE_M * 3) {
        int mm = m0 + tid / 3;
        mm = mm < M ? mm : M - 1;
        q_lds[tid] = out_points[mm * 3 + (tid % 3)];
    }
    if (tid >= 64 && tid < 64 + KP * 3) {
        kp_lds[tid - 64] = k_points[tid - 64];
    }
    __syncthreads();

    // ---------- Phase A: per wave, 4 output points.
    // Weights in VALU, then WF[m](16k x 64f) = W(16k x 32n) x G(32n x 64f) via WMMA f32 16x16x4.
    for (int mi = 0; mi < 4; ++mi) {
        const int m = wave * 4 + mi;
        int mm = m0 + m; mm = mm < M ? mm : M - 1;

        // lane <-> neighbor n
        const int gi = nbr_idx[mm * NNB + lane];
        idx_lds[wave * NNB + lane] = gi;
        const float px = points[gi * 3 + 0];
        const float py = points[gi * 3 + 1];
        const float pz = points[gi * 3 + 2];
        const float qx = q_lds[m * 3 + 0];
        const float qy = q_lds[m * 3 + 1];
        const float qz = q_lds[m * 3 + 2];
        #pragma unroll
        for (int k = 0; k < KP; ++k) {
            const float dx = px - qx - kp_lds[k * 3 + 0];
            const float dy = py - qy - kp_lds[k * 3 + 1];
            const float dz = pz - qz - kp_lds[k * 3 + 2];
            const float d  = sqrtf(dx * dx + dy * dy + dz * dz);
            W_lds[wave * WSTR + lane * 16 + k] = fmaxf(1.0f - d, 0.0f);   // EXTENT = 1.0
        }
        W_lds[wave * WSTR + lane * 16 + KP] = 0.0f;   // pad row k=15
        // per-wave LDS RAW: DS ops from one wave execute in order; no block barrier needed.

        #pragma unroll
        for (int ft = 0; ft < 4; ++ft) {
            v8f c = {};
            #pragma unroll
            for (int nc = 0; nc < 8; ++nc) {
                const int n0 = nc * 4 + 2 * half;     // A VGPR0/1: K = n0, n0+1 (per half)
                v2f a, b;
                a.x = W_lds[wave * WSTR + (n0 + 0) * 16 + l];
                a.y = W_lds[wave * WSTR + (n0 + 1) * 16 + l];
                const int i0 = idx_lds[wave * NNB + n0 + 0];
                const int i1 = idx_lds[wave * NNB + n0 + 1];
                b.x = features[i0 * FDIM + ft * 16 + l];   // B row n0, col f
                b.y = features[i1 * FDIM + ft * 16 + l];   // B row n0+1
                c = __builtin_amdgcn_wmma_f32_16x16x4_f32(
                        false, a, false, b, (short)0, c, false, false);
            }
            #pragma unroll
            for (int r = 0; r < 8; ++r) {
                const int k = r + 8 * half;           // D row
                if (k < KP)
                    WF_lds[m * WF_STRIDE + k * FDIM + ft * 16 + l] = c[r];
            }
        }
    }
    __syncthreads();

    // ---------- Phase B: out(16m x 64c) = WF(16m x 960) x KV(960 x 64).
    // Each wave owns one 16-wide c-tile; 240 K-chunks of 4, fp32 accumulation.
    const int ct = wave;
    v8f acc = {};
    #pragma unroll 4
    for (int j = 0; j < 240; ++j) {
        const int kf = j * 4 + 2 * half;              // K indices kf, kf+1 (per half)
        const v2f a = *reinterpret_cast<const v2f*>(&WF_lds[l * WF_STRIDE + kf]);
        v2f b;
        b.x = k_values[(kf + 0) * CDIM + ct * 16 + l];
        b.y = k_values[(kf + 1) * CDIM + ct * 16 + l];
        acc = __builtin_amdgcn_wmma_f32_16x16x4_f32(
                  false, a, false, b, (short)0, acc, false, false);
    }
    #pragma unroll
    for (int r = 0; r < 8; ++r) {
        const int row = m0 + r + 8 * half;
        if (row < M)
            out[row * CDIM + ct * 16 + l] = acc[r];
    }
}

extern "C" void kernel_launch(void* const* d_in, const int* in_sizes, int n_in,
                              void* d_out, int out_size, void* d_ws, size_t ws_size,
                              hipStream_t stream) {
    const float* points     = (const float*)d_in[0];
    const float* features   = (const float*)d_in[1];
    const float* out_points = (const float*)d_in[2];
    const int*   nbr_idx    = (const int*)d_in[3];
    const float* k_points   = (const float*)d_in[4];
    const float* k_values   = (const float*)d_in[5];
    float* out = (float*)d_out;

    const int M = in_sizes[3] / NNB;                  // neighbor_indices is [M,32]
    const int blocks = (M + TILE_M - 1) / TILE_M;     // 6250 for M=100000
    kpconv_wmma_kernel<<<blocks, 128, 0, stream>>>(
        points, features, out_points, nbr_idx, k_points, k_values, out, M);
    (void)d_ws; (void)ws_size; (void)out_size; (void)n_in;
}